// MemoryRamModule_25623774888598
// MI455X (gfx1250) — compile-verified
//
#include <hip/hip_runtime.h>
#include <hip/hip_bf16.h>
#include <math.h>

// ---------------------------------------------------------------------------
// Problem constants (match reference)
// ---------------------------------------------------------------------------
#define I_SZ   1024
#define H_SZ   512
#define M_BANK 100
#define M_PAD  112      // padded column stride for 100-wide activations
#define N_IMG  32768

typedef __attribute__((ext_vector_type(2))) float v2f;
typedef __attribute__((ext_vector_type(8))) float v8f;

// ---------------------------------------------------------------------------
// CDNA5 async copy helpers (ASYNCcnt-tracked global->LDS DMA, no VGPR transit)
// ---------------------------------------------------------------------------
__device__ __forceinline__ void async_copy_b128(void* lds_dst, const void* gsrc) {
    const unsigned lds_off = (unsigned)(size_t)lds_dst;   // addrspace(3) offset
    asm volatile("global_load_async_to_lds_b128 %0, %1, off"
                 :: "v"(lds_off), "v"(gsrc) : "memory");
}
__device__ __forceinline__ void wait_async0() {
    asm volatile("s_wait_asynccnt 0x0" ::: "memory");
}

// ---------------------------------------------------------------------------
// Kernel 1: f32 WMMA GEMM  out[N x ldout] = X[N x 1024] @ B[1024 x ncolsB]
//   256 threads = 8 waves (2x4); block tile 64 rows x 128 cols; each wave
//   owns a 32x32 C tile (4 f32 16x16 accumulators).  K staged 32 at a time
//   through double-buffered LDS filled by GLOBAL_LOAD_ASYNC_TO_LDS_B128;
//   copies for tile n+1 are in flight while tile n runs on the matrix pipe.
// ---------------------------------------------------------------------------
__global__ __launch_bounds__(256)
void gemm_f32_wmma(const float* __restrict__ X, const float* __restrict__ B,
                   float* __restrict__ out, int ncolsB, int ldout) {
    __shared__ __align__(16) float As[2][64][36];    // 64 rows x 32 k (stride 36)
    __shared__ __align__(16) float Bs[2][32][132];   // 32 k x 128 cols (stride 132)

    const int tid  = threadIdx.x;
    const int row0 = blockIdx.x * 64;
    const int col0 = blockIdx.y * 128;

    const int wave = tid >> 5;
    const int lane = tid & 31;
    const int wrow = wave >> 2;          // 0..1
    const int wcol = wave & 3;           // 0..3
    const int hi   = lane >> 4;          // lane half
    const int mn   = lane & 15;          // M for A-frag, N for B/C-frag

    // staging coordinates (per thread)
    const int ar  = tid >> 3;            // A: rows 0..31 (+32 on 2nd chunk)
    const int akq = (tid & 7) * 4;       // A: k offset, 16B chunk
    const int bkr = tid >> 5;            // B: k rows 0..7 (+8 per chunk)
    const int bc0 = (tid & 31) * 4;      // B: col offset, 16B chunk

    auto issue_tile = [&](int buf, int kk) {
        // A tile: 64x32 floats = 512 chunks, 2 per thread
        #pragma unroll
        for (int q = 0; q < 2; ++q) {
            const int r = ar + q * 32;
            async_copy_b128(&As[buf][r][akq],
                            X + (size_t)(row0 + r) * I_SZ + kk + akq);
        }
        // B tile: 32x128 floats = 1024 chunks, 4 per thread
        #pragma unroll
        for (int q = 0; q < 4; ++q) {
            const int kr = bkr + q * 8;
            async_copy_b128(&Bs[buf][kr][bc0],
                            B + (size_t)(kk + kr) * ncolsB + col0 + bc0);
        }
    };

    v8f acc[2][2] = {};

    issue_tile(0, 0);
    for (int kk = 0; kk < I_SZ; kk += 32) {
        const int cur = (kk >> 5) & 1;
        wait_async0();            // my share of copies into `cur` has landed
        __syncthreads();          // everyone's share landed; prev reads done
        if (kk + 32 < I_SZ) issue_tile(cur ^ 1, kk + 32);

        #pragma unroll
        for (int k4 = 0; k4 < 32; k4 += 4) {
            v2f a0, a1, b0, b1;
            // 16x4 f32 A fragments (two M tiles share the B fragments)
            a0.x = As[cur][wrow * 32 + mn][k4 + 2 * hi + 0];
            a0.y = As[cur][wrow * 32 + mn][k4 + 2 * hi + 1];
            a1.x = As[cur][wrow * 32 + 16 + mn][k4 + 2 * hi + 0];
            a1.y = As[cur][wrow * 32 + 16 + mn][k4 + 2 * hi + 1];
            // 4x16 f32 B fragments (two N tiles)
            b0.x = Bs[cur][k4 + 2 * hi + 0][wcol * 32 + mn];
            b0.y = Bs[cur][k4 + 2 * hi + 1][wcol * 32 + mn];
            b1.x = Bs[cur][k4 + 2 * hi + 0][wcol * 32 + 16 + mn];
            b1.y = Bs[cur][k4 + 2 * hi + 1][wcol * 32 + 16 + mn];
            acc[0][0] = __builtin_amdgcn_wmma_f32_16x16x4_f32(
                false, a0, false, b0, (short)0, acc[0][0], false, false);
            acc[0][1] = __builtin_amdgcn_wmma_f32_16x16x4_f32(
                false, a0, false, b1, (short)0, acc[0][1], false, false);
            acc[1][0] = __builtin_amdgcn_wmma_f32_16x16x4_f32(
                false, a1, false, b0, (short)0, acc[1][0], false, false);
            acc[1][1] = __builtin_amdgcn_wmma_f32_16x16x4_f32(
                false, a1, false, b1, (short)0, acc[1][1], false, false);
        }
        __syncthreads();          // done reading `cur` before it is refilled
    }

    // store C: within a 16x16 tile, vgpr v -> M = v + 8*hi ; N = lane&15
    #pragma unroll
    for (int mi = 0; mi < 2; ++mi) {
        #pragma unroll
        for (int ni = 0; ni < 2; ++ni) {
            const int gc = col0 + wcol * 32 + ni * 16 + mn;
            if (gc < ldout) {
                #pragma unroll
                for (int v = 0; v < 8; ++v) {
                    const int gr = row0 + wrow * 32 + mi * 16 + v + 8 * hi;
                    out[(size_t)gr * ldout + gc] = acc[mi][ni][v];
                }
            }
        }
    }
}

// ---------------------------------------------------------------------------
// Kernel 2: gate columns  Xwg[t] = X[t]·Wwg[0:1024], Xrg[t] = X[t]·Wrg[0:1024]
// ---------------------------------------------------------------------------
__global__ __launch_bounds__(256)
void gate_dots(const float* __restrict__ X, const float* __restrict__ Wwg,
               const float* __restrict__ Wrg, float* __restrict__ ow,
               float* __restrict__ org) {
    __shared__ float s1[256], s2[256];
    const int t = blockIdx.x, tid = threadIdx.x;
    float p1 = 0.f, p2 = 0.f;
    for (int i = tid; i < I_SZ; i += 256) {
        const float x = X[t * I_SZ + i];
        p1 += x * Wwg[i];
        p2 += x * Wrg[i];
    }
    s1[tid] = p1; s2[tid] = p2;
    __syncthreads();
    for (int o = 128; o > 0; o >>= 1) {
        if (tid < o) { s1[tid] += s1[tid + o]; s2[tid] += s2[tid + o]; }
        __syncthreads();
    }
    if (tid == 0) { ow[t] = s1[0]; org[t] = s2[0]; }
}

// ---------------------------------------------------------------------------
// Kernel 3: sequential scan. One persistent workgroup (1024 threads, 32 waves).
// mem (100x512 f32 = 200KB) and h live in LDS; recurrent weights stream from
// L2 (they total ~3.4MB, fully L2-resident).
// ---------------------------------------------------------------------------
__device__ __forceinline__ void softmax128(float* v, float* red, int tid) {
    // v[128]: entries 100..127 pre-set to -3.4e38 (mask); all threads call.
    if (tid < 128) red[tid] = v[tid];
    __syncthreads();
    for (int o = 64; o > 0; o >>= 1) {
        if (tid < o) red[tid] = fmaxf(red[tid], red[tid + o]);
        __syncthreads();
    }
    const float mx = red[0];
    __syncthreads();
    if (tid < 128) { const float e = __expf(v[tid] - mx); v[tid] = e; red[tid] = e; }
    __syncthreads();
    for (int o = 64; o > 0; o >>= 1) {
        if (tid < o) red[tid] += red[tid + o];
        __syncthreads();
    }
    const float inv = 1.0f / red[0];
    __syncthreads();
    if (tid < 128) v[tid] *= inv;
    __syncthreads();
}

__global__ __launch_bounds__(1024)
void scan_kernel(const float* __restrict__ Wc,  const float* __restrict__ bc,
                 const float* __restrict__ Wwg, const float* __restrict__ bwg,
                 const float* __restrict__ Wwp, const float* __restrict__ bwp,
                 const float* __restrict__ Wrg, const float* __restrict__ brg,
                 const float* __restrict__ Wrp, const float* __restrict__ brp,
                 const float* __restrict__ Wrh, const float* __restrict__ Whh,
                 const float* __restrict__ bh,
                 const float* __restrict__ Xxh, const float* __restrict__ Xc,
                 const float* __restrict__ Xwp, const float* __restrict__ Xrp,
                 const float* __restrict__ Xwg, const float* __restrict__ Xrg,
                 float* __restrict__ out) {
    extern __shared__ float smem[];
    float* memS = smem;               // 51200
    float* hS   = memS + 51200;       // 512
    float* cS   = hS   + 512;         // 512
    float* rS   = cS   + 512;         // 512
    float* part = rS   + 512;         // 1024
    float* lr   = part + 1024;        // 128  (read logits -> ar)
    float* lw   = lr   + 128;         // 128  (write logits -> aw)
    float* wgr  = lw   + 128;         // 512  (Wrg h-part, cached)
    float* wgw  = wgr  + 512;         // 512  (Wwg h-part, cached)
    float* sred = wgw  + 512;         // 8    (broadcast scalars)

    const int tid = threadIdx.x;

    // h-part weight base pointers
    const float* WrpH = Wrp + I_SZ * M_BANK;
    const float* WwpH = Wwp + I_SZ * M_BANK;
    const float* WcH  = Wc  + I_SZ * H_SZ;

    // init state
    for (int e = tid; e < M_BANK * H_SZ; e += 1024) memS[e] = 0.0f;
    if (tid < 512) {
        hS[tid]  = 0.0f;
        wgr[tid] = Wrg[I_SZ + tid];
        wgw[tid] = Wwg[I_SZ + tid];
    }
    __syncthreads();

    for (int t = 0; t < N_IMG; ++t) {
        // ---- (1) read-projection logits: lr[j] = Xrp + brp + h·WrpH[:,j]
        {
            const int j = tid & 127, s = tid >> 7;   // 8 slices of 64 rows
            float p = 0.f;
            if (j < M_BANK) {
                const float* col = WrpH + (s * 64) * M_BANK + j;
                #pragma unroll 8
                for (int i = 0; i < 64; ++i) p += hS[s * 64 + i] * col[i * M_BANK];
            }
            part[tid] = p;
            __syncthreads();
            if (tid < M_BANK) {
                float v = Xrp[t * M_PAD + tid] + brp[tid];
                #pragma unroll
                for (int s2 = 0; s2 < 8; ++s2) v += part[tid + 128 * s2];
                lr[tid] = v;
            } else if (tid < 128) {
                lr[tid] = -3.4e38f;
            }
            __syncthreads();
        }
        // ---- (2) write-projection logits: lw[j]
        {
            const int j = tid & 127, s = tid >> 7;
            float p = 0.f;
            if (j < M_BANK) {
                const float* col = WwpH + (s * 64) * M_BANK + j;
                #pragma unroll 8
                for (int i = 0; i < 64; ++i) p += hS[s * 64 + i] * col[i * M_BANK];
            }
            part[tid] = p;
            __syncthreads();
            if (tid < M_BANK) {
                float v = Xwp[t * M_PAD + tid] + bwp[tid];
                #pragma unroll
                for (int s2 = 0; s2 < 8; ++s2) v += part[tid + 128 * s2];
                lw[tid] = v;
            } else if (tid < 128) {
                lw[tid] = -3.4e38f;
            }
            __syncthreads();
        }
        // ---- (3) gate dots (both gates in one pass over LDS-cached columns)
        {
            part[tid] = (tid < 512) ? hS[tid] * wgr[tid]
                                    : hS[tid - 512] * wgw[tid - 512];
            __syncthreads();
            for (int o = 256; o > 0; o >>= 1) {
                if (tid < o) part[tid] += part[tid + o];
                else if (tid >= 512 && tid < 512 + o) part[tid] += part[tid + o];
                __syncthreads();
            }
            if (tid == 0) {
                const float glr = part[0]   + Xrg[t] + brg[0];
                const float glw = part[512] + Xwg[t] + bwg[0];
                sred[0] = 1.0f / (1.0f + __expf(-glr));   // go
                sred[1] = 1.0f / (1.0f + __expf(-glw));   // gw
            }
            __syncthreads();
        }
        // ---- (4) softmaxes: lr -> ar, lw -> aw
        softmax128(lr, part, tid);
        softmax128(lw, part, tid);

        // ---- (5) r = go * (ar @ mem)
        {
            const int j = tid & 511, half = tid >> 9;
            float p = 0.f;
            const int m0 = half * 50;
            #pragma unroll 10
            for (int m = 0; m < 50; ++m) p += lr[m0 + m] * memS[(m0 + m) * H_SZ + j];
            part[tid] = p;
            __syncthreads();
            if (tid < 512) rS[tid] = sred[0] * (part[tid] + part[tid + 512]);
            __syncthreads();
        }
        // ---- (6) c = relu(Xc + bc + h @ WcH)
        {
            const int j = tid & 511, half = tid >> 9;
            const float* col = WcH + (half * 256) * H_SZ + j;
            float p = 0.f;
            #pragma unroll 8
            for (int i = 0; i < 256; ++i) p += hS[half * 256 + i] * col[i * H_SZ];
            part[tid] = p;
            __syncthreads();
            if (tid < 512) {
                const float v = Xc[t * H_SZ + tid] + bc[tid] + part[tid] + part[tid + 512];
                cS[tid] = fmaxf(v, 0.0f);
            }
            __syncthreads();
        }
        // ---- (7) h_new = relu(Xxh + bh + r @ Wrh + h @ Whh); overwrite h, emit
        {
            const int j = tid & 511, half = tid >> 9;
            const float* ch = Whh + (half * 256) * H_SZ + j;
            const float* cr = Wrh + (half * 256) * H_SZ + j;
            float p = 0.f;
            #pragma unroll 4
            for (int i = 0; i < 256; ++i) {
                p += hS[half * 256 + i] * ch[i * H_SZ];
                p += rS[half * 256 + i] * cr[i * H_SZ];
            }
            part[tid] = p;
            __syncthreads();
            if (tid < 512) {
                float hv = Xxh[t * H_SZ + tid] + bh[tid] + part[tid] + part[tid + 512];
                hv = fmaxf(hv, 0.0f);
                hS[tid] = hv;
                out[t * H_SZ + tid] = hv;
            }
            __syncthreads();
        }
        // ---- (8) mem = gw*aw[:,None]*c[None,:] + (1-aw[:,None])*mem
        {
            const float gw = sred[1];
            for (int e = tid; e < M_BANK * H_SZ; e += 1024) {
                const int m = e >> 9, j = e & 511;
                const float a = lw[m];
                memS[e] = gw * a * cS[j] + (1.0f - a) * memS[e];
            }
            __syncthreads();
        }
    }
}

// ---------------------------------------------------------------------------
// Host launcher
// ---------------------------------------------------------------------------
extern "C" void kernel_launch(void* const* d_in, const int* in_sizes, int n_in,
                              void* d_out, int out_size, void* d_ws, size_t ws_size,
                              hipStream_t stream) {
    const float* X   = (const float*)d_in[0];   // hidden_frames (N, 1024)
    const float* Wc  = (const float*)d_in[1];   // (1536, 512)
    const float* bc  = (const float*)d_in[2];
    const float* Wwg = (const float*)d_in[3];   // (1536, 1)
    const float* bwg = (const float*)d_in[4];
    const float* Wwp = (const float*)d_in[5];   // (1536, 100)
    const float* bwp = (const float*)d_in[6];
    const float* Wrg = (const float*)d_in[7];   // (1536, 1)
    const float* brg = (const float*)d_in[8];
    const float* Wrp = (const float*)d_in[9];   // (1536, 100)
    const float* brp = (const float*)d_in[10];
    const float* Wxh = (const float*)d_in[11];  // (1024, 512)
    const float* Wrh = (const float*)d_in[12];  // (512, 512)
    const float* Whh = (const float*)d_in[13];  // (512, 512)
    const float* bh  = (const float*)d_in[14];
    float* out = (float*)d_out;                 // (N, 512)

    // workspace layout (floats)
    float* ws   = (float*)d_ws;
    float* Xxh  = ws;                              // N*512
    float* Xc   = Xxh + (size_t)N_IMG * H_SZ;      // N*512
    float* Xwp  = Xc  + (size_t)N_IMG * H_SZ;      // N*112
    float* Xrp  = Xwp + (size_t)N_IMG * M_PAD;     // N*112
    float* Xwg  = Xrp + (size_t)N_IMG * M_PAD;     // N
    float* Xrg  = Xwg + (size_t)N_IMG;             // N

    // --- x-side precompute: async-pipelined f32 WMMA GEMMs
    dim3 blk(256);
    gemm_f32_wmma<<<dim3(N_IMG / 64, H_SZ / 128), blk, 0, stream>>>(X, Wxh, Xxh, H_SZ, H_SZ);
    gemm_f32_wmma<<<dim3(N_IMG / 64, H_SZ / 128), blk, 0, stream>>>(X, Wc,  Xc,  H_SZ, H_SZ);
    gemm_f32_wmma<<<dim3(N_IMG / 64, 1),          blk, 0, stream>>>(X, Wwp, Xwp, M_BANK, M_PAD);
    gemm_f32_wmma<<<dim3(N_IMG / 64, 1),          blk, 0, stream>>>(X, Wrp, Xrp, M_BANK, M_PAD);
    gate_dots<<<dim3(N_IMG), blk, 0, stream>>>(X, Wwg, Wrg, Xwg, Xrg);

    // --- sequential scan: single persistent workgroup, mem+h in LDS
    const size_t lds_bytes = (size_t)(51200 + 512 + 512 + 512 + 1024 +
                                      128 + 128 + 512 + 512 + 8) * sizeof(float);
    scan_kernel<<<dim3(1), dim3(1024), lds_bytes, stream>>>(
        Wc, bc, Wwg, bwg, Wwp, bwp, Wrg, brg, Wrp, brp,
        Wrh, Whh, bh, Xxh, Xc, Xwp, Xrp, Xwg, Xrg, out);
}